// Actor_56916906607124
// MI455X (gfx1250) — compile-verified
//
#include <hip/hip_runtime.h>
#include <stdint.h>

// ---------------------------------------------------------------------------
// compute_encoder_mask for B=8, N=2048, K=16 on MI455X (gfx1250).
// One 256-thread workgroup (8 wave32s) per output row. Row data (dist, twc,
// depot) is staged global->LDS with CDNA5 async-to-LDS ops (ASYNCcnt path),
// selection of the 16 nearest neighbors uses register-resident 64-bit keys +
// wave32 shuffle butterflies, and rows with depot[b][i]==1 take a store-only
// fast path (no reads at all).
// ---------------------------------------------------------------------------

#ifndef __has_builtin
#define __has_builtin(x) 0
#endif

#if defined(__HIP_DEVICE_COMPILE__) && __has_builtin(__builtin_amdgcn_global_load_async_to_lds_b128)
#define ASYNC_LDS 1
#endif

#define NT  256   // threads per block (8 waves of 32)
#define PER 8     // elements per thread (N / NT)
#define MAXN 2048

typedef unsigned long long u64;
typedef unsigned int u32;
typedef int v4i __attribute__((ext_vector_type(4)));

#ifdef ASYNC_LDS
typedef __attribute__((address_space(1))) v4i* g_v4i_p;   // global int4*
typedef __attribute__((address_space(3))) v4i* l_v4i_p;   // LDS int4*

__device__ __forceinline__ void async_cp16(const void* g, void* l) {
    __builtin_amdgcn_global_load_async_to_lds_b128((g_v4i_p)g, (l_v4i_p)l, 0, 0);
}
#endif

__device__ __forceinline__ u64 u64min(u64 a, u64 b) { return a < b ? a : b; }

__global__ __launch_bounds__(NT)
void encoder_mask_kernel(const float* __restrict__ dist,
                         const float* __restrict__ maxd,   // [B]
                         const int*   __restrict__ twc,
                         const int*   __restrict__ depot,  // [B,N]
                         const int*   __restrict__ kptr,   // scalar K
                         float*       __restrict__ out,
                         int N)
{
    const int row = blockIdx.x;          // b*N + i
    const int b   = row / N;
    const int i   = row - b * N;
    const int tid = threadIdx.x;

    const float* drow = dist  + (size_t)row * N;
    const int*   trow = twc   + (size_t)row * N;
    const int*   prow = depot + (size_t)b   * N;
    float*       orow = out   + (size_t)row * N;

    // ---- Fast path: depot[b][i] == 1 -> entire row is ones (uniform branch).
    if (prow[i] != 0) {
        const float4 ones = {1.0f, 1.0f, 1.0f, 1.0f};
        float4* o4 = (float4*)orow;
        #pragma unroll
        for (int u = 0; u < PER / 4; ++u)
            o4[tid * (PER / 4) + u] = ones;
        return;
    }

    __shared__ float s_dist[MAXN];
    __shared__ int   s_twc[MAXN];
    __shared__ int   s_dep[MAXN];
    __shared__ unsigned char s_sel[MAXN];
    __shared__ u64   s_part[NT / 32];
    __shared__ u64   s_winner;

    // ---- Stage row data into LDS. On CDNA5 use async global->LDS DMA ops
    //      (tracked by ASYNCcnt), which never touch VGPRs.
#ifdef ASYNC_LDS
    for (int c = tid; c < N / 4; c += NT) {
        async_cp16(drow + 4 * c, &s_dist[4 * c]);
        async_cp16(trow + 4 * c, &s_twc[4 * c]);
        async_cp16(prow + 4 * c, &s_dep[4 * c]);
    }
#if __has_builtin(__builtin_amdgcn_s_wait_asynccnt)
    __builtin_amdgcn_s_wait_asynccnt(0);
#else
    asm volatile("s_wait_asynccnt 0" ::: "memory");
#endif
#else
    for (int c = tid; c < N / 4; c += NT) {
        ((float4*)s_dist)[c] = ((const float4*)drow)[c];
        ((int4*)s_twc)[c]    = ((const int4*)trow)[c];
        ((int4*)s_dep)[c]    = ((const int4*)prow)[c];
    }
#endif
    for (int c = tid; c < N / 4; c += NT)
        ((u32*)s_sel)[c] = 0;
    __syncthreads();

    // ---- Build per-thread 64-bit sort keys in registers.
    //      key = (float_bits << 32) | j : non-negative IEEE bits are order
    //      preserving, and the embedded index reproduces jax top_k's
    //      lowest-index tie-break (incl. among blocked max_dist*10 entries).
    int k = kptr[0];
    if (k > N) k = N;
    if (k < 0) k = 0;
    const u32 pushedbits = __float_as_uint(maxd[b] * 10.0f);

    u64 kreg[PER];
    #pragma unroll
    for (int u = 0; u < PER; ++u) {
        const int j = tid + NT * u;                    // stride-NT: LDS conflict-free
        const bool blocked = (s_twc[j] == 0) || (j == i);
        const u32 kb = blocked ? pushedbits : __float_as_uint(s_dist[j]);
        kreg[u] = ((u64)kb << 32) | (u32)j;
    }

    // ---- Extract the k smallest keys (16 rounds of block-wide argmin).
    for (int it = 0; it < k; ++it) {
        u64 best = ~0ull;
        #pragma unroll
        for (int u = 0; u < PER; ++u) best = u64min(best, kreg[u]);

        // wave32 butterfly min
        #pragma unroll
        for (int off = 16; off > 0; off >>= 1)
            best = u64min(best, (u64)__shfl_xor((unsigned long long)best, off, 32));

        if ((tid & 31) == 0) s_part[tid >> 5] = best;
        __syncthreads();

        if (tid == 0) {
            u64 w = s_part[0];
            #pragma unroll
            for (int wv = 1; wv < NT / 32; ++wv) w = u64min(w, s_part[wv]);
            s_winner = w;
            s_sel[(u32)(w & 0xFFFFFFFFull)] = 1;
        }
        __syncthreads();

        // consume the winner (keys are unique -> exactly one lane matches)
        const u64 w = s_winner;
        #pragma unroll
        for (int u = 0; u < PER; ++u)
            if (kreg[u] == w) kreg[u] = ~0ull;
    }

    // ---- Emit: (selected & twc) | depot[j] | (i==j)   (depot[i]==0 here).
    #pragma unroll
    for (int u0 = 0; u0 < PER; u0 += 4) {
        const int j0 = tid * PER + u0;
        float4 v;
        float* vv = (float*)&v;
        #pragma unroll
        for (int q = 0; q < 4; ++q) {
            const int j = j0 + q;
            const bool on = (s_sel[j] && (s_twc[j] != 0)) || (s_dep[j] != 0) || (j == i);
            vv[q] = on ? 1.0f : 0.0f;
        }
        ((float4*)orow)[j0 >> 2] = v;
    }
}

extern "C" void kernel_launch(void* const* d_in, const int* in_sizes, int n_in,
                              void* d_out, int out_size, void* d_ws, size_t ws_size,
                              hipStream_t stream) {
    (void)n_in; (void)out_size; (void)d_ws; (void)ws_size;

    const float* dist  = (const float*)d_in[0];
    const float* maxd  = (const float*)d_in[1];
    const int*   twc   = (const int*)d_in[2];
    const int*   depot = (const int*)d_in[3];
    const int*   kptr  = (const int*)d_in[4];
    float*       out   = (float*)d_out;

    const int BN = in_sizes[3];          // B*N
    const int N  = in_sizes[0] / BN;     // N   (2048)
    // one workgroup per output row
    encoder_mask_kernel<<<dim3(BN), dim3(NT), 0, stream>>>(dist, maxd, twc, depot, kptr, out, N);
}